// LSTMWithPositiveAndNormedWeights_42056319762970
// MI455X (gfx1250) — compile-verified
//
#include <hip/hip_runtime.h>

typedef __attribute__((ext_vector_type(2))) float v2f;
typedef __attribute__((ext_vector_type(4))) float v4f;
typedef __attribute__((ext_vector_type(8))) float v8f;

#define B_   4096
#define T_   256
#define F_   64
#define H_   16
#define G4H  64
#define D_   64
#define XPAD 68    // x row stride (floats): even -> 8B-aligned b64 reads, conflict-free
#define HPAD 18    // h row stride (floats): even, 18n mod 64 distinct for n<16
#define WPP  160   // swizzled W pair-row stride (floats): 160 mod 64 == 32 -> halves on disjoint banks

__device__ __forceinline__ float sigf(float v) { return 1.0f / (1.0f + __expf(-v)); }

__global__ __launch_bounds__(32)
void lstm_fused_kernel(const float* __restrict__ x, const float* __restrict__ W,
                       const float* __restrict__ U, const float* __restrict__ b,
                       const float* __restrict__ Wd, const float* __restrict__ bd,
                       float* __restrict__ out) {
  // W swizzled as K-pairs: Wlds[pp*WPP + col*2 + j] = W[(2*pp+j)*64 + col], pp=0..31
  __shared__ float Wlds[32 * WPP];          // 20 KB
  __shared__ float Wdlds[8 * WPP];          // 5 KB, same layout, pp=0..7
  __shared__ float xlds[2][16 * XPAD];      // double-buffered x tile
  __shared__ float hlds[16 * HPAD];         // h staging (C-layout -> A-layout)

  const int lane  = threadIdx.x & 31;
  const int n     = lane & 15;          // N index (B/C/D) or M index (A)
  const int half  = lane >> 4;          // 0 / 1
  const int khalf = half * 2;           // K sub-offset within a k-step
  const int mhalf = half * 8;           // C/D row offset for upper half-wave
  const int b0    = blockIdx.x * 16;

  // Stage W / Wd pair-swizzled (one-time)
  for (int i = lane; i < F_ * G4H; i += 32) {
    int j = i & 1, col = (i >> 1) & 63, pp = i >> 7;
    Wlds[pp * WPP + col * 2 + j] = W[(2 * pp + j) * G4H + col];
  }
  for (int i = lane; i < H_ * D_; i += 32) {
    int j = i & 1, col = (i >> 1) & 63, pp = i >> 7;
    Wdlds[pp * WPP + col * 2 + j] = Wd[(2 * pp + j) * D_ + col];
  }
  __syncthreads();   // single-wave workgroup: lowers to S_NOP, LDS is in-order per wave

  // --- U fragments live in registers (serial h@U critical path) ---
  v2f ufrag[4][4];
#pragma unroll
  for (int kk = 0; kk < 4; ++kk)
#pragma unroll
    for (int g = 0; g < 4; ++g) {
      int k0 = kk * 4 + khalf;
      ufrag[kk][g].x = U[k0 * G4H + g * 16 + n];
      ufrag[kk][g].y = U[(k0 + 1) * G4H + g * 16 + n];
    }

  float bg[4];
#pragma unroll
  for (int g = 0; g < 4; ++g) bg[g] = b[g * 16 + n];

  v8f h = {}; v8f c = {};
  float* xl0 = &xlds[0][0];
  float* xl1 = &xlds[1][0];
  float* hl  = &hlds[0];

  auto loadx = [&](int t, v4f* xr) {
#pragma unroll
    for (int i = 0; i < 8; ++i) {
      int idx = i * 32 + lane;
      int row = idx >> 4, c4 = idx & 15;
      xr[i] = *(const v4f*)(x + ((size_t)(b0 + row) * T_ + t) * F_ + c4 * 4);
    }
  };
  auto storex = [&](float* dst, const v4f* xr) {
#pragma unroll
    for (int i = 0; i < 8; ++i) {
      int idx = i * 32 + lane;
      int row = idx >> 4, c4 = idx & 15;
      float* p = dst + row * XPAD + c4 * 4;
      v2f lo; lo.x = xr[i].x; lo.y = xr[i].y;
      v2f hi; hi.x = xr[i].z; hi.y = xr[i].w;
      *(v2f*)p = lo;
      *(v2f*)(p + 2) = hi;
    }
  };

  v4f xr[8];
  loadx(0, xr);
  storex(xl0, xr);

  int p = 0;
  for (int t = 0; t < T_; ++t) {
    if (t + 1 < T_) loadx(t + 1, xr);          // prefetch next tile into registers
    if (t + 2 < T_) {                          // nudge t+2 toward L2 (global_prefetch_b8)
      const float* pf = x + ((size_t)(b0 + (lane >> 1)) * T_ + (t + 2)) * F_ + (lane & 1) * 32;
      __builtin_prefetch(pf, 0, 0);
    }

    // z accumulators, one 16x16 tile per gate (i,f,g,o), init with bias
    v8f acc[4];
#pragma unroll
    for (int g = 0; g < 4; ++g)
#pragma unroll
      for (int r = 0; r < 8; ++r) acc[g][r] = bg[g];

    float* xl = p ? xl1 : xl0;

    // x_t @ W : K=64 -> 16 k-steps x 4 gates; B-fragment = one ds_load_b64
#pragma unroll 4
    for (int kk = 0; kk < 16; ++kk) {
      v2f a = *(const v2f*)(xl + n * XPAD + kk * 4 + khalf);
      const float* wrow = Wlds + (kk * 2 + half) * WPP;
#pragma unroll
      for (int g = 0; g < 4; ++g) {
        v2f wb = *(const v2f*)(wrow + (g * 16 + n) * 2);
        acc[g] = __builtin_amdgcn_wmma_f32_16x16x4_f32(false, a, false, wb,
                                                       (short)0, acc[g], false, false);
      }
    }

    // h @ U : round-trip h (C-layout) through LDS to rebuild A-layout fragments
#pragma unroll
    for (int r = 0; r < 8; ++r) hl[(r + mhalf) * HPAD + n] = h[r];
#pragma unroll
    for (int kk = 0; kk < 4; ++kk) {
      v2f a = *(const v2f*)(hl + n * HPAD + kk * 4 + khalf);
#pragma unroll
      for (int g = 0; g < 4; ++g)
        acc[g] = __builtin_amdgcn_wmma_f32_16x16x4_f32(false, a, false, ufrag[kk][g],
                                                       (short)0, acc[g], false, false);
    }

    // gates: i,f sigmoid; g relu; o sigmoid; h = o * relu(c)
#pragma unroll
    for (int r = 0; r < 8; ++r) {
      float gi = sigf(acc[0][r]);
      float gf = sigf(acc[1][r]);
      float gg = fmaxf(acc[2][r], 0.0f);
      float go = sigf(acc[3][r]);
      float cn = gf * c[r] + gi * gg;
      c[r] = cn;
      h[r] = go * fmaxf(cn, 0.0f);
    }

    if (t + 1 < T_) { storex(p ? xl0 : xl1, xr); p ^= 1; }
  }

  // ---- Dense head: sigmoid(h @ Wd + bd), then row-normalize ----
#pragma unroll
  for (int r = 0; r < 8; ++r) hl[(r + mhalf) * HPAD + n] = h[r];

  v8f dacc[4];
#pragma unroll
  for (int j = 0; j < 4; ++j) {
    float bj = bd[j * 16 + n];
#pragma unroll
    for (int r = 0; r < 8; ++r) dacc[j][r] = bj;
  }
#pragma unroll
  for (int kk = 0; kk < 4; ++kk) {
    v2f a = *(const v2f*)(hl + n * HPAD + kk * 4 + khalf);
    const float* wrow = Wdlds + (kk * 2 + half) * WPP;
#pragma unroll
    for (int j = 0; j < 4; ++j) {
      v2f wb = *(const v2f*)(wrow + (j * 16 + n) * 2);
      dacc[j] = __builtin_amdgcn_wmma_f32_16x16x4_f32(false, a, false, wb,
                                                      (short)0, dacc[j], false, false);
    }
  }

  float* ol = xl0;  // reuse x buffer as the 16x64 output staging tile
#pragma unroll
  for (int j = 0; j < 4; ++j)
#pragma unroll
    for (int r = 0; r < 8; ++r)
      ol[(r + mhalf) * XPAD + j * 16 + n] = sigf(dacc[j][r]);

  if (lane < 16) {   // per-row inverse sums (rows 0..15), same-wave LDS is in-order
    float s = 0.0f;
    for (int q = 0; q < 64; ++q) s += ol[lane * XPAD + q];
    hl[lane] = 1.0f / s;
  }

#pragma unroll 4
  for (int i = 0; i < 32; ++i) {
    int idx = i * 32 + lane;
    int row = idx >> 6, col = idx & 63;
    out[(size_t)(b0 + row) * D_ + col] = ol[row * XPAD + col] * hl[row];
  }
}

extern "C" void kernel_launch(void* const* d_in, const int* in_sizes, int n_in,
                              void* d_out, int out_size, void* d_ws, size_t ws_size,
                              hipStream_t stream) {
  (void)in_sizes; (void)n_in; (void)out_size; (void)d_ws; (void)ws_size;
  const float* x  = (const float*)d_in[0];
  const float* W  = (const float*)d_in[1];
  const float* U  = (const float*)d_in[2];
  const float* b  = (const float*)d_in[3];
  const float* Wd = (const float*)d_in[4];
  const float* bd = (const float*)d_in[5];
  float* out = (float*)d_out;

  dim3 grid(B_ / 16);   // 256 single-wave blocks -> 256 independent recurrence streams
  dim3 block(32);       // one wave32 per block
  hipLaunchKernelGGL(lstm_fused_kernel, grid, block, 0, stream,
                     x, W, U, b, Wd, bd, out);
}